// Hyena1D_76141180223625
// MI455X (gfx1250) — compile-verified
//
#include <hip/hip_runtime.h>
#include <hip/hip_bf16.h>

typedef __attribute__((ext_vector_type(2))) float v2f;
typedef __attribute__((ext_vector_type(8))) float v8f;

#define PI_F 3.14159265358979323846f

// D = A(16x4) * B(4x16) + C, f32 WMMA (CDNA5 V_WMMA_F32_16X16X4_F32)
static __device__ __forceinline__ v8f wmma4(v2f a, v2f b, v8f c) {
  return __builtin_amdgcn_wmma_f32_16x16x4_f32(false, a, false, b, (short)0, c,
                                               false, false);
}

// Build DFT-64 cos/sin tables in LDS: table[r*64+k] = cos/sin(2*pi*r*k/64)
static __device__ __forceinline__ void initF(float* sCos, float* sSin) {
  for (int i = threadIdx.x; i < 64 * 64; i += blockDim.x) {
    int r = i >> 6, k = i & 63;
    float th = (2.f * PI_F / 64.f) * (float)((r * k) & 63);  // exact period reduction
    float s, c;
    __sincosf(th, &s, &c);
    sCos[i] = c;
    sSin[i] = s;
  }
  __syncthreads();
}

// ---------------- tiny filter MLP:  Hbins[2049][1024] ----------------
__global__ void __launch_bounds__(256) kH(const float* __restrict__ W1,
                                          const float* __restrict__ b1,
                                          const float* __restrict__ W2,
                                          const float* __restrict__ b2,
                                          float* __restrict__ Hb) {
  __shared__ float h[32];
  const int bin = blockIdx.x;
  const int tid = threadIdx.x;
  if (tid < 32) {
    float acc = b1[tid];
#pragma unroll
    for (int i = 0; i < 16; ++i) {
      int m = i >> 1;
      float invf = expf((float)(2 * m) * (-9.210340371976184f / 16.f));
      float arg = (float)bin * invf;
      float pe = (i & 1) ? cosf(arg) : sinf(arg);
      acc += pe * W1[i * 32 + tid];
    }
    h[tid] = acc / (1.f + expf(-acc));  // SiLU
  }
  __syncthreads();
  int d = blockIdx.y * 256 + tid;
  float acc = b2[d];
#pragma unroll
  for (int j = 0; j < 32; ++j) acc += h[j] * W2[j * 1024 + d];
  Hb[(size_t)bin * 1024 + d] = acc;
}

// ---------------- Kernel A: stage-1 forward DFT over t1 + twiddle ----------------
// G[b][k1][t0][d] = w4096^{t0*k1} * sum_t1 e^{-2pi i k1 t1/64} x[b][64*t1+t0][d]
__global__ void __launch_bounds__(128) kA(const float* __restrict__ x,
                                          float* __restrict__ Br,
                                          float* __restrict__ Bi) {
  __shared__ float sCos[4096], sSin[4096];
  initF(sCos, sSin);
  const int lane = threadIdx.x & 31;
  const int wave = threadIdx.x >> 5;
  const int half = lane >> 4;
  const int n16 = lane & 15;
  const int d0 = blockIdx.x * 16;
  const int t0 = blockIdx.y * 4 + wave;
  const int b = blockIdx.z;

  const float* xb = x + (size_t)b * 4096 * 1024 + d0 + n16;

  v2f bx[16];  // real input B-fragments, K = t1
#pragma unroll
  for (int kk = 0; kk < 16; ++kk) {
    int r0 = 4 * kk + 2 * half;
    bx[kk].x = xb[(size_t)(64 * r0 + t0) * 1024];
    bx[kk].y = xb[(size_t)(64 * (r0 + 1) + t0) * 1024];
  }

#pragma unroll
  for (int m = 0; m < 4; ++m) {
    v8f accR = {0, 0, 0, 0, 0, 0, 0, 0};
    v8f accI = {0, 0, 0, 0, 0, 0, 0, 0};
    const int mrow = 16 * m + n16;
#pragma unroll
    for (int kk = 0; kk < 16; ++kk) {
      int i0 = mrow * 64 + 4 * kk + 2 * half;
      v2f aR, aI;
      aR.x = sCos[i0];
      aR.y = sCos[i0 + 1];
      aI.x = -sSin[i0];  // forward: Fi = -sin
      aI.y = -sSin[i0 + 1];
      accR = wmma4(aR, bx[kk], accR);
      accI = wmma4(aI, bx[kk], accI);
    }
#pragma unroll
    for (int v = 0; v < 8; ++v) {
      int M = 16 * m + v + 8 * half;  // k1
      float s, c;
      __sincosf(-(2.f * PI_F / 4096.f) * (float)((t0 * M) & 4095), &s, &c);
      float gr = accR[v] * c - accI[v] * s;
      float gi = accR[v] * s + accI[v] * c;
      size_t o = (((size_t)b * 64 + M) * 64 + t0) * 1024 + d0 + n16;
      Br[o] = gr;
      Bi[o] = gi;
    }
  }
}

// ---------------- Kernel B: fwd over t0, *Hfull, inv over k2, conj twiddle (in place) ----
__global__ void __launch_bounds__(128) kB(float* __restrict__ Br,
                                          float* __restrict__ Bi,
                                          const float* __restrict__ Hb) {
  __shared__ float sCos[4096], sSin[4096];
  __shared__ float sXr[4][1024], sXi[4][1024];
  initF(sCos, sSin);
  const int lane = threadIdx.x & 31;
  const int wave = threadIdx.x >> 5;
  const int half = lane >> 4;
  const int n16 = lane & 15;
  const int d0 = blockIdx.x * 16;
  const int k1 = blockIdx.y * 4 + wave;
  const int b = blockIdx.z;
  float* xr = sXr[wave];
  float* xi = sXi[wave];

  const size_t base = (((size_t)b * 64 + k1) * 64) * 1024 + d0 + n16;  // + t0*1024

  v2f bR[16], bI[16];  // G[t0][d], K = t0
#pragma unroll
  for (int kk = 0; kk < 16; ++kk) {
    int r0 = 4 * kk + 2 * half;
    bR[kk].x = Br[base + (size_t)r0 * 1024];
    bR[kk].y = Br[base + (size_t)(r0 + 1) * 1024];
    bI[kk].x = Bi[base + (size_t)r0 * 1024];
    bI[kk].y = Bi[base + (size_t)(r0 + 1) * 1024];
  }

  // stage 1: Xf[k2][d] = sum_t0 e^{-2pi i k2 t0/64} G[t0][d], then * Hfull[k1+64*k2][d]
#pragma unroll
  for (int m = 0; m < 4; ++m) {
    v8f accR = {0, 0, 0, 0, 0, 0, 0, 0};
    v8f accI = {0, 0, 0, 0, 0, 0, 0, 0};
    const int mrow = 16 * m + n16;
#pragma unroll
    for (int kk = 0; kk < 16; ++kk) {
      int i0 = mrow * 64 + 4 * kk + 2 * half;
      float c0 = sCos[i0], c1 = sCos[i0 + 1];
      float s0 = sSin[i0], s1 = sSin[i0 + 1];
      v2f aR, aI, aN;
      aR.x = c0;  aR.y = c1;
      aI.x = -s0; aI.y = -s1;  // Fi
      aN.x = s0;  aN.y = s1;   // -Fi
      accR = wmma4(aR, bR[kk], accR);
      accR = wmma4(aN, bI[kk], accR);
      accI = wmma4(aR, bI[kk], accI);
      accI = wmma4(aI, bR[kk], accI);
    }
#pragma unroll
    for (int v = 0; v < 8; ++v) {
      int k2 = 16 * m + v + 8 * half;
      int k = k1 + 64 * k2;
      int bin = (k <= 2048) ? k : (4096 - k);  // real symmetric filter
      float hv = Hb[(size_t)bin * 1024 + d0 + n16];
      xr[k2 * 16 + n16] = accR[v] * hv;
      xi[k2 * 16 + n16] = accI[v] * hv;
    }
  }

  // stage 2: C[t0][d] = sum_k2 e^{+2pi i t0 k2/64} Xf[k2][d], then * e^{+2pi i t0 k1/4096}
#pragma unroll
  for (int m = 0; m < 4; ++m) {
    v8f accR = {0, 0, 0, 0, 0, 0, 0, 0};
    v8f accI = {0, 0, 0, 0, 0, 0, 0, 0};
    const int mrow = 16 * m + n16;
#pragma unroll
    for (int kk = 0; kk < 16; ++kk) {
      int i0 = mrow * 64 + 4 * kk + 2 * half;
      float c0 = sCos[i0], c1 = sCos[i0 + 1];
      float s0 = sSin[i0], s1 = sSin[i0 + 1];
      v2f aR, aI, aN;
      aR.x = c0;  aR.y = c1;
      aI.x = s0;  aI.y = s1;    // Fi = +sin (inverse)
      aN.x = -s0; aN.y = -s1;   // -Fi
      int r0 = 4 * kk + 2 * half;
      v2f vR, vI;
      vR.x = xr[r0 * 16 + n16];
      vR.y = xr[(r0 + 1) * 16 + n16];
      vI.x = xi[r0 * 16 + n16];
      vI.y = xi[(r0 + 1) * 16 + n16];
      accR = wmma4(aR, vR, accR);
      accR = wmma4(aN, vI, accR);
      accI = wmma4(aR, vI, accI);
      accI = wmma4(aI, vR, accI);
    }
#pragma unroll
    for (int v = 0; v < 8; ++v) {
      int t0 = 16 * m + v + 8 * half;
      float s, c;
      __sincosf((2.f * PI_F / 4096.f) * (float)((t0 * k1) & 4095), &s, &c);
      float gr = accR[v] * c - accI[v] * s;
      float gi = accR[v] * s + accI[v] * c;
      size_t o = base + (size_t)t0 * 1024;
      Br[o] = gr;
      Bi[o] = gi;
    }
  }
}

// ---------------- Kernel C: inverse stage-2 over k1, real part only ----------------
// y[b][64*t1+t0][d] = (1/4096)*gain[d]*Re( sum_k1 e^{+2pi i t1 k1/64} C[k1][t0][d] )
__global__ void __launch_bounds__(128) kC(const float* __restrict__ Br,
                                          const float* __restrict__ Bi,
                                          const float* __restrict__ gain,
                                          float* __restrict__ y) {
  __shared__ float sCos[4096], sSin[4096];
  initF(sCos, sSin);
  const int lane = threadIdx.x & 31;
  const int wave = threadIdx.x >> 5;
  const int half = lane >> 4;
  const int n16 = lane & 15;
  const int d0 = blockIdx.x * 16;
  const int t0 = blockIdx.y * 4 + wave;
  const int b = blockIdx.z;

  const size_t base =
      ((size_t)b * 64) * 64 * 1024 + (size_t)t0 * 1024 + d0 + n16;  // + k1*64*1024

  v2f bR[16], bI[16];  // C[k1][d], K = k1
#pragma unroll
  for (int kk = 0; kk < 16; ++kk) {
    int r0 = 4 * kk + 2 * half;
    bR[kk].x = Br[base + (size_t)r0 * 64 * 1024];
    bR[kk].y = Br[base + (size_t)(r0 + 1) * 64 * 1024];
    bI[kk].x = Bi[base + (size_t)r0 * 64 * 1024];
    bI[kk].y = Bi[base + (size_t)(r0 + 1) * 64 * 1024];
  }

  const float ag = gain[d0 + n16] * (1.f / 4096.f);

#pragma unroll
  for (int m = 0; m < 4; ++m) {
    v8f accR = {0, 0, 0, 0, 0, 0, 0, 0};
    const int mrow = 16 * m + n16;
#pragma unroll
    for (int kk = 0; kk < 16; ++kk) {
      int i0 = mrow * 64 + 4 * kk + 2 * half;
      v2f aR, aN;
      aR.x = sCos[i0];
      aR.y = sCos[i0 + 1];
      aN.x = -sSin[i0];  // -Fi, Fi = +sin (inverse)
      aN.y = -sSin[i0 + 1];
      accR = wmma4(aR, bR[kk], accR);  // Re = cos*Cr - sin*Ci
      accR = wmma4(aN, bI[kk], accR);
    }
#pragma unroll
    for (int v = 0; v < 8; ++v) {
      int t1 = 16 * m + v + 8 * half;
      int t = 64 * t1 + t0;
      y[((size_t)b * 4096 + t) * 1024 + d0 + n16] = accR[v] * ag;
    }
  }
}

extern "C" void kernel_launch(void* const* d_in, const int* in_sizes, int n_in,
                              void* d_out, int out_size, void* d_ws, size_t ws_size,
                              hipStream_t stream) {
  const float* x = (const float*)d_in[0];
  const float* W1 = (const float*)d_in[1];
  const float* b1 = (const float*)d_in[2];
  const float* W2 = (const float*)d_in[3];
  const float* b2 = (const float*)d_in[4];
  const float* ag = (const float*)d_in[5];
  float* out = (float*)d_out;

  // workspace layout: Hbins[2049*1024] | Br[4*4096*1024] | Bi[4*4096*1024]  (~143 MB)
  float* Hb = (float*)d_ws;
  float* wBr = Hb + (size_t)2049 * 1024;
  float* wBi = wBr + (size_t)4 * 4096 * 1024;

  kH<<<dim3(2049, 4, 1), 256, 0, stream>>>(W1, b1, W2, b2, Hb);
  kA<<<dim3(64, 16, 4), 128, 0, stream>>>(x, wBr, wBi);
  kB<<<dim3(64, 16, 4), 128, 0, stream>>>(wBr, wBi, Hb);
  kC<<<dim3(64, 16, 4), 128, 0, stream>>>(wBr, wBi, ag, out);
}